// AdditiveAttention_36532991820318
// MI455X (gfx1250) — compile-verified
//
#include <hip/hip_runtime.h>
#include <math.h>

typedef __attribute__((ext_vector_type(2))) float v2f;
typedef __attribute__((ext_vector_type(8))) float v8f;

// async-to-LDS builtin pointer types (param type per hipcc diagnostic:
// "int __attribute__((vector_size(16))) addrspace(1)*")
typedef int v4i_vs __attribute__((vector_size(16)));
typedef __attribute__((address_space(1))) v4i_vs* gasync_p;
typedef __attribute__((address_space(3))) v4i_vs* lasync_p;

#define BB 4
#define LQ 256
#define LK 512
#define HH 256

// Native CDNA5 transcendental tanh (V_TANH_F32) when available.
__device__ __forceinline__ float aa_tanh(float x) {
#if __has_builtin(__builtin_amdgcn_tanhf)
    return __builtin_amdgcn_tanhf(x);
#else
    return tanhf(x);
#endif
}

// ---------------------------------------------------------------------------
// Kernel 1: Y[M,H] = X[M,H] @ W[H,H]^T + bias   (nn.Linear convention)
// One 16x16 output tile per wave, V_WMMA_F32_16X16X4_F32, K stepped by 4.
// ---------------------------------------------------------------------------
__global__ __launch_bounds__(256)
void aa_proj_kernel(const float* __restrict__ X, const float* __restrict__ W,
                    const float* __restrict__ bias, float* __restrict__ Y,
                    int Mtotal) {
    const int lane  = threadIdx.x & 31;
    const int wave  = (blockIdx.x * 256 + threadIdx.x) >> 5;
    const int tilesN = HH / 16;                 // 16
    const int tileM  = wave / tilesN;
    const int tileN  = wave % tilesN;
    if (tileM * 16 >= Mtotal) return;           // wave-uniform guard

    const int m0   = tileM * 16;
    const int n0   = tileN * 16;
    const int l15  = lane & 15;
    const int koff = (lane >> 4) * 2;           // 0 or 2

    const float* Arow = X + (m0 + l15) * HH + koff;   // row of X
    const float* Brow = W + (n0 + l15) * HH + koff;   // row of W (== col of W^T)

    v8f acc = {};
    #pragma unroll 4
    for (int h0 = 0; h0 < HH; h0 += 4) {
        v2f a = *(const v2f*)(Arow + h0);
        v2f b = *(const v2f*)(Brow + h0);
        acc = __builtin_amdgcn_wmma_f32_16x16x4_f32(
            false, a, false, b, (short)0, acc, false, false);
    }

    const float bn    = bias[n0 + l15];
    const int   mbase = m0 + ((lane >> 4) << 3);      // +8 for upper half-wave
    #pragma unroll
    for (int j = 0; j < 8; ++j)
        Y[(mbase + j) * HH + (n0 + l15)] = acc[j] + bn;
}

// ---------------------------------------------------------------------------
// Kernel 2: fused additive scores + masked softmax, one workgroup per (b, qi).
// scores[k] = bv + sum_h tanh(qp[h] + kp[k][h]) * v[h]; softmax over k.
// kp streamed through LDS in 256-row x 32-col panels, pitch 36 words:
//   - 144-byte rows => every float4 chunk is 16B-aligned (async b128 legal)
//   - gcd(36,64)=4 => only a 2-way bank conflict on compute-side ds reads,
//     irrelevant against the tanh(trans-pipe)-limited inner loop.
// Panels filled with GLOBAL_LOAD_ASYNC_TO_LDS_B128 (ASYNCcnt) when available.
// ---------------------------------------------------------------------------
#define KPITCH 36

__global__ __launch_bounds__(256)
void aa_score_softmax_kernel(const float* __restrict__ qp,
                             const float* __restrict__ kp,
                             const float* __restrict__ vvec,
                             const float* __restrict__ bvp,
                             const int*   __restrict__ mask,
                             float* __restrict__ weights /* [B,LQ,LK] */) {
    const int b   = blockIdx.x / LQ;
    const int qi  = blockIdx.x % LQ;
    const int tid = threadIdx.x;

    __shared__ float qs[HH];
    __shared__ float vs[HH];
    __shared__ float kpan[256 * KPITCH];   // 36 KB panel
    __shared__ float sc[LK];
    __shared__ float red[256];

    qs[tid] = qp[(b * LQ + qi) * HH + tid];
    vs[tid] = vvec[tid];

    for (int c = 0; c < 2; ++c) {                      // two blocks of 256 keys
        const float* kbase = kp + (size_t)(b * LK + c * 256) * HH;
        float acc = 0.0f;
        for (int hp = 0; hp < HH; hp += 32) {
            __syncthreads();                           // kpan reuse safety
            // cooperative panel fill: 2048 float4s, 8 per thread
            #pragma unroll
            for (int i = 0; i < 8; ++i) {
                const int f   = tid + 256 * i;         // float4 index
                const int row = f >> 3;
                const int c4  = (f & 7) << 2;
                const float* gp = kbase + row * HH + hp + c4;
                if (hp + 32 < HH)                      // warm next panel in L1
                    __builtin_prefetch(gp + 32, 0, 0); // -> global_prefetch_b8
                float* dst = &kpan[row * KPITCH + c4]; // 16B-aligned
#if __has_builtin(__builtin_amdgcn_global_load_async_to_lds_b128)
                __builtin_amdgcn_global_load_async_to_lds_b128(
                    (gasync_p)(void*)gp, (lasync_p)(void*)dst, 0, 0);
#else
                *(float4*)dst = *(const float4*)gp;
#endif
            }
#if __has_builtin(__builtin_amdgcn_global_load_async_to_lds_b128)
#if __has_builtin(__builtin_amdgcn_s_wait_asynccnt)
            __builtin_amdgcn_s_wait_asynccnt(0);
#else
            asm volatile("s_wait_asynccnt 0x0" ::: "memory");
#endif
#endif
            __syncthreads();
            const float* krow = &kpan[tid * KPITCH];
            #pragma unroll
            for (int cc = 0; cc < 32; ++cc)            // v_tanh_f32 + v_fma_f32
                acc = fmaf(aa_tanh(krow[cc] + qs[hp + cc]), vs[hp + cc], acc);
        }
        sc[c * 256 + tid] = acc;
    }
    __syncthreads();

    // masked softmax over 512 scores (2 per thread)
    const float bv = bvp[0];
    const int*  mb = mask + b * LK;
    const int   ma = mb[tid], mc = mb[tid + 256];
    float s0 = sc[tid] + bv;
    float s1 = sc[tid + 256] + bv;
    if (!ma) s0 = -INFINITY;
    if (!mc) s1 = -INFINITY;

    red[tid] = fmaxf(s0, s1);
    __syncthreads();
    #pragma unroll
    for (int s = 128; s > 0; s >>= 1) {
        if (tid < s) red[tid] = fmaxf(red[tid], red[tid + s]);
        __syncthreads();
    }
    const float mx = red[0];
    __syncthreads();

    const float e0 = ma ? __expf(s0 - mx) : 0.0f;      // v_exp_f32
    const float e1 = mc ? __expf(s1 - mx) : 0.0f;
    red[tid] = e0 + e1;
    __syncthreads();
    #pragma unroll
    for (int s = 128; s > 0; s >>= 1) {
        if (tid < s) red[tid] += red[tid + s];
        __syncthreads();
    }
    const float inv = 1.0f / red[0];

    float* wrow = weights + (size_t)(b * LQ + qi) * LK;
    wrow[tid]       = e0 * inv;
    wrow[tid + 256] = e1 * inv;
}

// ---------------------------------------------------------------------------
// Kernel 3: O[b,q,h] = sum_k W[b,q,k] * V[b,k,h]  via fp32 WMMA.
// ---------------------------------------------------------------------------
__global__ __launch_bounds__(256)
void aa_pv_kernel(const float* __restrict__ Wt /* [B,LQ,LK] */,
                  const float* __restrict__ V  /* [B,LK,H]  */,
                  float* __restrict__ O        /* [B,LQ,H]  */) {
    const int lane = threadIdx.x & 31;
    const int wave = (blockIdx.x * 256 + threadIdx.x) >> 5;
    const int tilesN = HH / 16;                 // 16
    const int tilesM = LQ / 16;                 // 16
    const int b   = wave / (tilesM * tilesN);
    const int rem = wave % (tilesM * tilesN);
    const int m0  = (rem / tilesN) * 16;
    const int n0  = (rem % tilesN) * 16;

    const float* Wb = Wt + (size_t)b * LQ * LK;
    const float* Vb = V  + (size_t)b * LK * HH;
    const int l15   = lane & 15;
    const int khalf = (lane >> 4) * 2;          // 0 or 2

    const float* Arow = Wb + (m0 + l15) * LK + khalf;

    v8f acc = {};
    #pragma unroll 4
    for (int k0 = 0; k0 < LK; k0 += 4) {
        v2f a = *(const v2f*)(Arow + k0);       // W[m][k0+khalf .. +1]
        v2f bvals;
        bvals.x = Vb[(k0 + khalf)     * HH + n0 + l15];
        bvals.y = Vb[(k0 + khalf + 1) * HH + n0 + l15];
        acc = __builtin_amdgcn_wmma_f32_16x16x4_f32(
            false, a, false, bvals, (short)0, acc, false, false);
    }

    const int mbase = m0 + ((lane >> 4) << 3);
    #pragma unroll
    for (int j = 0; j < 8; ++j)
        O[((size_t)b * LQ + mbase + j) * HH + (n0 + l15)] = acc[j];
}

// ---------------------------------------------------------------------------
extern "C" void kernel_launch(void* const* d_in, const int* in_sizes, int n_in,
                              void* d_out, int out_size, void* d_ws, size_t ws_size,
                              hipStream_t stream) {
    const float* query = (const float*)d_in[0];
    const float* key   = (const float*)d_in[1];
    const float* value = (const float*)d_in[2];
    const int*   mask  = (const int*)  d_in[3];
    const float* Wq    = (const float*)d_in[4];
    const float* bq    = (const float*)d_in[5];
    const float* Wk    = (const float*)d_in[6];
    const float* bk    = (const float*)d_in[7];
    const float* vv    = (const float*)d_in[8];
    const float* bv    = (const float*)d_in[9];

    float* qp  = (float*)d_ws;                        // [B,LQ,H] : 1 MB
    float* kpw = qp + (size_t)BB * LQ * HH;           // [B,LK,H] : 2 MB

    float* attn_out = (float*)d_out;                  // [B,LQ,H]
    float* attn_w   = attn_out + (size_t)BB * LQ * HH;// [B,LQ,LK]

    // 1) projections (fp32 WMMA)
    {
        const int tilesQ = (BB * LQ / 16) * (HH / 16);    // 1024 waves
        aa_proj_kernel<<<tilesQ / 8, 256, 0, stream>>>(query, Wq, bq, qp, BB * LQ);
        const int tilesK = (BB * LK / 16) * (HH / 16);    // 2048 waves
        aa_proj_kernel<<<tilesK / 8, 256, 0, stream>>>(key, Wk, bk, kpw, BB * LK);
    }

    // 2) fused tanh-scores + masked softmax -> attn_weights
    aa_score_softmax_kernel<<<BB * LQ, 256, 0, stream>>>(qp, kpw, vv, bv, mask, attn_w);

    // 3) attn_output = attn_weights @ value (fp32 WMMA)
    {
        const int tilesO = BB * (LQ / 16) * (HH / 16);    // 1024 waves
        aa_pv_kernel<<<tilesO / 8, 256, 0, stream>>>(attn_w, value, attn_out);
    }
}